// PGExplainer_48661979464123
// MI455X (gfx1250) — compile-verified
//
#include <hip/hip_runtime.h>
#include <math.h>

typedef __attribute__((ext_vector_type(2))) float v2f;
typedef __attribute__((ext_vector_type(8))) float v8f;

#define DFEAT 128
#define HID   64
#define GTEMP 5.0f
#define GBIAS 0.0001f

// D = A(16x4 f32) * B(4x16 f32) + C(16x16 f32)
#define WMMA_F32_16x16x4(a, b, c) \
  __builtin_amdgcn_wmma_f32_16x16x4_f32(false, (a), false, (b), (short)0, (c), false, false)

// -------- Gumbel-sigmoid gate, shared by edge kernels --------
__device__ __forceinline__ float pgx_gate(float sw, float uu) {
  float eps = (GBIAS - (1.0f - GBIAS)) * uu + (1.0f - GBIAS);  // U(bias, 1-bias)
  float g   = logf(eps) - log1pf(-eps);
  float z   = (g + sw) * (1.0f / GTEMP);
  return 1.0f / (1.0f + expf(-z));
}

// -------- Kernel 0: cvec[h] = b1[h] + embeds[n,:] . W1[256+k, h] --------
__global__ void pgx_cvec(const float* __restrict__ embeds,
                         const float* __restrict__ W1,
                         const float* __restrict__ b1,
                         const int*   __restrict__ nptr,
                         float* __restrict__ cvec) {
  int h = threadIdx.x;                      // 64 threads
  int nn = nptr[0];
  const float* erow = embeds + (size_t)nn * DFEAT;
  float s = b1[h];
#pragma unroll 4
  for (int k = 0; k < DFEAT; ++k)
    s = fmaf(erow[k], W1[(size_t)(2 * DFEAT + k) * HID + h], s);
  cvec[h] = s;
}

// -------- Kernel 1: node-level GEMMs via fp32 WMMA --------
// A = embeds[N,128] @ W1[0:128,64];  B = embeds @ W1[128:256,64]
// One wave per 16-node tile; 8 f32 accumulators (2 parts x 4 N-tiles),
// K swept in steps of 4 (V_WMMA_F32_16X16X4_F32). The A-fragment is shared
// between the two GEMMs (same left matrix), so one b64 load feeds 8 WMMAs.
__global__ __launch_bounds__(128)
void pgx_node_gemm(const float* __restrict__ embeds,
                   const float* __restrict__ W1,
                   float* __restrict__ Atab,
                   float* __restrict__ Btab,
                   int nNodes, int nTiles) {
  int wave = blockIdx.x * (blockDim.x >> 5) + (threadIdx.x >> 5);
  if (wave >= nTiles) return;               // wave-uniform exit: EXEC stays all-1 for WMMA
  int lane = threadIdx.x & 31;
  int base = wave * 16;
  int m    = lane & 15;                     // A-matrix row (M), also B column (N)
  int kh   = (lane >> 4) * 2;               // lanes 16-31 hold K=2,3

  int mrow = base + m; if (mrow >= nNodes) mrow = nNodes - 1;   // clamp (dup rows harmless)
  const float* arow   = embeds + (size_t)mrow * DFEAT + kh;
  const float* w0base = W1 + (size_t)kh * HID + m;              // part 0: rows 0..127
  const float* w1base = W1 + (size_t)(DFEAT + kh) * HID + m;    // part 1: rows 128..255

  v8f accA[4] = {};
  v8f accB[4] = {};

  for (int kk = 0; kk < DFEAT; kk += 4) {
    v2f a = *(const v2f*)(arow + kk);       // K = kk+kh, kk+kh+1
    const float* w0 = w0base + (size_t)kk * HID;
    const float* w1 = w1base + (size_t)kk * HID;
#pragma unroll
    for (int nt = 0; nt < 4; ++nt) {
      v2f b0; b0.x = w0[nt * 16]; b0.y = w0[HID + nt * 16];
      accA[nt] = WMMA_F32_16x16x4(a, b0, accA[nt]);
      v2f b1v; b1v.x = w1[nt * 16]; b1v.y = w1[HID + nt * 16];
      accB[nt] = WMMA_F32_16x16x4(a, b1v, accB[nt]);
    }
  }

  // C/D layout: VGPR r -> M = r (lanes 0-15) / r+8 (lanes 16-31); N = lane&15
  int nlo = lane & 15;
  int mo  = (lane >> 4) * 8;
#pragma unroll
  for (int nt = 0; nt < 4; ++nt) {
#pragma unroll
    for (int r = 0; r < 8; ++r) {
      int mm = base + mo + r;
      if (mm < nNodes) {
        size_t off = (size_t)mm * HID + nt * 16 + nlo;
        Atab[off] = accA[nt][r];
        Btab[off] = accB[nt][r];
      }
    }
  }
}

// -------- Kernel 2: edge pass. 8 lanes per edge, float4 gathers from L2 --------
__global__ __launch_bounds__(256)
void pgx_edge(const float* __restrict__ Atab, const float* __restrict__ Btab,
              const float* __restrict__ cvec, const float* __restrict__ W2,
              const float* __restrict__ b2,   const float* __restrict__ u,
              const int*   __restrict__ src,  const int* __restrict__ dst,
              float* __restrict__ out, int E) {
  int t   = blockIdx.x * blockDim.x + threadIdx.x;
  int e   = t >> 3;
  int sub = t & 7;
  if (e >= E) return;
  int s  = src[e];
  int d2 = dst[e];
  const float4* ap = (const float4*)(Atab + (size_t)s  * HID + sub * 8);
  const float4* bp = (const float4*)(Btab + (size_t)d2 * HID + sub * 8);
  const float4* cp = (const float4*)(cvec + sub * 8);
  const float4* wp = (const float4*)(W2   + sub * 8);
  float acc = 0.0f;
#pragma unroll
  for (int i = 0; i < 2; ++i) {
    float4 a = ap[i], b = bp[i], c = cp[i], w = wp[i];
    float h0 = a.x + b.x + c.x; h0 = h0 > 0.0f ? h0 : 0.0f;
    float h1 = a.y + b.y + c.y; h1 = h1 > 0.0f ? h1 : 0.0f;
    float h2 = a.z + b.z + c.z; h2 = h2 > 0.0f ? h2 : 0.0f;
    float h3 = a.w + b.w + c.w; h3 = h3 > 0.0f ? h3 : 0.0f;
    acc = fmaf(h0, w.x, acc);
    acc = fmaf(h1, w.y, acc);
    acc = fmaf(h2, w.z, acc);
    acc = fmaf(h3, w.w, acc);
  }
  acc += __shfl_xor(acc, 1);
  acc += __shfl_xor(acc, 2);
  acc += __shfl_xor(acc, 4);
  if (sub == 0) {
    out[e] = pgx_gate(acc + b2[0], u[e]);
  }
}

// -------- Fallback: monolithic per-edge compute (used only if ws too small) --------
__global__ __launch_bounds__(256)
void pgx_edge_fallback(const float* __restrict__ embeds, const float* __restrict__ W1,
                       const float* __restrict__ b1,     const float* __restrict__ W2,
                       const float* __restrict__ b2,     const float* __restrict__ u,
                       const int*   __restrict__ src,    const int* __restrict__ dst,
                       const int*   __restrict__ nptr,   float* __restrict__ out, int E) {
  int t   = blockIdx.x * blockDim.x + threadIdx.x;
  int e   = t >> 3;
  int sub = t & 7;
  if (e >= E) return;
  int nn = nptr[0];
  int s  = src[e];
  int d2 = dst[e];
  const float* es = embeds + (size_t)s  * DFEAT;
  const float* ed = embeds + (size_t)d2 * DFEAT;
  const float* en = embeds + (size_t)nn * DFEAT;
  float hh[8];
#pragma unroll
  for (int j = 0; j < 8; ++j) hh[j] = b1[sub * 8 + j];
  for (int k = 0; k < DFEAT; ++k) {
    float evs = es[k], evd = ed[k], evn = en[k];
    const float* wa = W1 + (size_t)k * HID + sub * 8;
    const float* wb = wa + (size_t)DFEAT * HID;
    const float* wc = wb + (size_t)DFEAT * HID;
#pragma unroll
    for (int j = 0; j < 8; ++j)
      hh[j] = fmaf(evs, wa[j], fmaf(evd, wb[j], fmaf(evn, wc[j], hh[j])));
  }
  float acc = 0.0f;
#pragma unroll
  for (int j = 0; j < 8; ++j) {
    float hv = hh[j] > 0.0f ? hh[j] : 0.0f;
    acc = fmaf(hv, W2[sub * 8 + j], acc);
  }
  acc += __shfl_xor(acc, 1);
  acc += __shfl_xor(acc, 2);
  acc += __shfl_xor(acc, 4);
  if (sub == 0) {
    out[e] = pgx_gate(acc + b2[0], u[e]);
  }
}

extern "C" void kernel_launch(void* const* d_in, const int* in_sizes, int n_in,
                              void* d_out, int out_size, void* d_ws, size_t ws_size,
                              hipStream_t stream) {
  const float* embeds = (const float*)d_in[0];
  const float* W1     = (const float*)d_in[1];
  const float* b1     = (const float*)d_in[2];
  const float* W2     = (const float*)d_in[3];
  const float* b2     = (const float*)d_in[4];
  const float* u      = (const float*)d_in[5];
  const int*   src    = (const int*)d_in[6];
  const int*   dst    = (const int*)d_in[7];
  const int*   nptr   = (const int*)d_in[8];

  int E = in_sizes[6];
  int N = in_sizes[0] / DFEAT;
  float* out = (float*)d_out;

  size_t need = ((size_t)2 * N * HID + HID) * sizeof(float);
  int edgeBlocks = (E * 8 + 255) / 256;

  if (ws_size >= need) {
    float* Atab = (float*)d_ws;
    float* Btab = Atab + (size_t)N * HID;
    float* cvec = Btab + (size_t)N * HID;

    pgx_cvec<<<1, HID, 0, stream>>>(embeds, W1, b1, nptr, cvec);

    int tiles   = (N + 15) / 16;
    int blocks1 = (tiles + 3) / 4;              // 4 waves / block
    pgx_node_gemm<<<blocks1, 128, 0, stream>>>(embeds, W1, Atab, Btab, N, tiles);

    pgx_edge<<<edgeBlocks, 256, 0, stream>>>(Atab, Btab, cvec, W2, b2, u,
                                             src, dst, out, E);
  } else {
    pgx_edge_fallback<<<edgeBlocks, 256, 0, stream>>>(embeds, W1, b1, W2, b2, u,
                                                      src, dst, nptr, out, E);
  }
}